// Spatial_Attn_67894843015614
// MI455X (gfx1250) — compile-verified
//
#include <hip/hip_runtime.h>
#include <hip/hip_bf16.h>

// ---------------- problem constants (from setup_inputs) ----------------
#define Bb   4
#define Vv   3
#define BV   12          // B*V
#define Cc   512
#define C2   1024        // 2*C
#define Nt   1920        // H*W = 24*80
#define ROWCHUNKS 16     // attn column-sum partials (1920/16 = 120 rows each)
#define BN_EPS 1e-5f

typedef float v2f __attribute__((ext_vector_type(2)));
typedef float v8f __attribute__((ext_vector_type(8)));
typedef int   v4i __attribute__((ext_vector_type(4)));

#define AS1 __attribute__((address_space(1)))
#define AS3 __attribute__((address_space(3)))
typedef AS1 v4i* gv4i_p;   // global-memory pointer to 16B packet
typedef AS3 v4i* lv4i_p;   // LDS pointer to 16B packet

#if defined(__has_builtin)
#if __has_builtin(__builtin_amdgcn_global_load_async_to_lds_b128) && \
    __has_builtin(__builtin_amdgcn_s_wait_asynccnt)
#define HAVE_ASYNC_LDS 1
#endif
#endif
#ifndef HAVE_ASYNC_LDS
#define HAVE_ASYNC_LDS 0
#endif

// ---------------- K1: attn = exp(-dist/(2*sigma)), partial col-sums, attention out
__global__ __launch_bounds__(128) void attn_kernel(
    const float* __restrict__ dist, const float* __restrict__ sigma,
    float* __restrict__ attn, float* __restrict__ partial,
    float* __restrict__ attnOut)
{
    const int m     = blockIdx.x * 128 + threadIdx.x;   // column
    const int bv    = blockIdx.y;
    const int chunk = blockIdx.z;
    const float inv2s = 1.0f / (2.0f * sigma[0]);
    const int rows = Nt / ROWCHUNKS;
    const int n0 = chunk * rows;
    const size_t base = (size_t)bv * Nt * Nt;
    const int v = bv % Vv;
    float* aout = (v == 0) ? (attnOut + (size_t)(bv / Vv) * Nt * Nt) : nullptr;

    float acc = 0.0f;
    for (int i = 0; i < rows; ++i) {
        const int n = n0 + i;
        const size_t idx = base + (size_t)n * Nt + m;
        const float e = __expf(-dist[idx] * inv2s);
        attn[idx] = e;
        if (aout) aout[(size_t)n * Nt + m] = e;
        acc += e;
    }
    partial[((size_t)chunk * BV + bv) * Nt + m] = acc;
}

// ---------------- K2: deterministic reduction of column-sum partials
__global__ __launch_bounds__(256) void norm_fin(
    const float* __restrict__ partial, float* __restrict__ norm)
{
    const int idx = blockIdx.x * 256 + threadIdx.x;     // over BV*Nt
    float s = 0.0f;
    #pragma unroll
    for (int ch = 0; ch < ROWCHUNKS; ++ch)
        s += partial[(size_t)ch * (BV * Nt) + idx];
    norm[idx] = s;
}

// ---------------- K3: fp32 WMMA GEMM  Out[o][n] = sum_k A[o][k]*B[k][n] (+bias[o]) (*1/colDen[n])
#define BMT 64
#define BNT 64
#define BKT 16
#define LDA_S 20    // padded LDS stride for A tile (conflict-free; 80B rows keep 16B alignment)
#define LDB_S 68    // padded LDS stride for B tile (272B rows keep 16B alignment)

__device__ __forceinline__ void wmma_compute_stage(
    const float* __restrict__ AsB, const float* __restrict__ BsB,
    int m0, int n0, int hl, int hi, v8f& acc0, v8f& acc1)
{
    #pragma unroll
    for (int kk = 0; kk < BKT; kk += 4) {
        const int ka = kk + hi * 2;
        v2f a, b0, b1;
        // A 16x4 fragment: lanes 0-15 -> K={kk,kk+1} of row m0+hl; lanes 16-31 -> K={kk+2,kk+3}
        a.x = AsB[(m0 + hl) * LDA_S + ka];
        a.y = AsB[(m0 + hl) * LDA_S + ka + 1];
        // B 4x16 fragments (two N halves), same half-wave K split
        b0.x = BsB[(ka    ) * LDB_S + n0 + hl];
        b0.y = BsB[(ka + 1) * LDB_S + n0 + hl];
        b1.x = BsB[(ka    ) * LDB_S + n0 + 16 + hl];
        b1.y = BsB[(ka + 1) * LDB_S + n0 + 16 + hl];
        acc0 = __builtin_amdgcn_wmma_f32_16x16x4_f32(false, a, false, b0, (short)0, acc0, false, false);
        acc1 = __builtin_amdgcn_wmma_f32_16x16x4_f32(false, a, false, b1, (short)0, acc1, false, false);
    }
}

__global__ __launch_bounds__(256) void wmma_gemm_f32(
    const float* __restrict__ A,  size_t aBv, int lda,
    const float* __restrict__ Bm, size_t bBv, int ldb,
    const float* __restrict__ bias,
    const float* __restrict__ colDen, size_t cBv,
    float* __restrict__ Out, size_t oBv, int ldo,
    int K)
{
    __shared__ float As[2][BMT * LDA_S];
    __shared__ float Bs[2][BKT * LDB_S];

    const int bv = blockIdx.z;
    A   += (size_t)bv * aBv;
    Bm  += (size_t)bv * bBv;
    Out += (size_t)bv * oBv;

    const int tileN = blockIdx.x * BNT;
    const int tileM = blockIdx.y * BMT;
    const int tid  = threadIdx.x;
    const int lane = tid & 31;
    const int wave = tid >> 5;          // 0..7
    const int hl   = lane & 15;
    const int hi   = (lane >> 4) & 1;   // half-wave select
    const int m0   = (wave & 3) * 16;   // wave's M subtile
    const int n0   = (wave >> 2) * 32;  // wave's N slab (two 16-wide halves)

    v8f acc0 = {0.f,0.f,0.f,0.f,0.f,0.f,0.f,0.f};
    v8f acc1 = {0.f,0.f,0.f,0.f,0.f,0.f,0.f,0.f};

    // cooperative tile load mapping (16B per thread per tile)
    const int arow = tid >> 2;          // 0..63
    const int acol = (tid & 3) * 4;     // 0,4,8,12
    const int brow = tid >> 4;          // 0..15
    const int bcol = (tid & 15) * 4;    // 0..60

    const float* Aptr = A  + (size_t)(tileM + arow) * lda + acol;   // + kt
    const float* Bptr = Bm + (size_t)brow * ldb + tileN + bcol;     // + kt*ldb
    float* AsDst0 = &As[0][arow * LDA_S + acol];
    float* BsDst0 = &Bs[0][brow * LDB_S + bcol];
    float* AsDst1 = &As[1][arow * LDA_S + acol];
    float* BsDst1 = &Bs[1][brow * LDB_S + bcol];

#if HAVE_ASYNC_LDS
    // ---- double-buffered async DMA staging (ASYNCcnt-tracked, L2 -> LDS direct) ----
    __builtin_amdgcn_global_load_async_to_lds_b128((gv4i_p)(Aptr), (lv4i_p)(AsDst0), 0, 0);
    __builtin_amdgcn_global_load_async_to_lds_b128((gv4i_p)(Bptr), (lv4i_p)(BsDst0), 0, 0);
    int cur = 0;
    for (int kt = 0; kt < K; kt += BKT) {
        const bool more = (kt + BKT) < K;
        if (more) {
            float* ad = cur ? AsDst0 : AsDst1;
            float* bd = cur ? BsDst0 : BsDst1;
            __builtin_amdgcn_global_load_async_to_lds_b128(
                (gv4i_p)(Aptr + (kt + BKT)), (lv4i_p)(ad), 0, 0);
            __builtin_amdgcn_global_load_async_to_lds_b128(
                (gv4i_p)(Bptr + (size_t)(kt + BKT) * ldb), (lv4i_p)(bd), 0, 0);
            __builtin_amdgcn_s_wait_asynccnt(2);   // current stage's 2 DMAs landed
        } else {
            __builtin_amdgcn_s_wait_asynccnt(0);
        }
        __syncthreads();                            // all waves' stage data visible
        wmma_compute_stage(&As[cur][0], &Bs[cur][0], m0, n0, hl, hi, acc0, acc1);
        __syncthreads();                            // buffer free before next DMA overwrites it
        cur ^= 1;
    }
#else
    // ---- register-staged double buffering: next tile loads overlap WMMA compute ----
    float4 av = *(const float4*)(Aptr);
    float4 bw = *(const float4*)(Bptr);
    int cur = 0;
    for (int kt = 0; kt < K; kt += BKT) {
        float* ad = cur ? AsDst1 : AsDst0;
        float* bd = cur ? BsDst1 : BsDst0;
        *(float4*)ad = av;
        *(float4*)bd = bw;
        __syncthreads();
        if (kt + BKT < K) {
            av = *(const float4*)(Aptr + (kt + BKT));
            bw = *(const float4*)(Bptr + (size_t)(kt + BKT) * ldb);
        }
        wmma_compute_stage(&As[cur][0], &Bs[cur][0], m0, n0, hl, hi, acc0, acc1);
        cur ^= 1;
    }
#endif

    // epilogue: C/D layout -> VGPR r holds M = r (lanes 0-15) / M = 8+r (lanes 16-31), N = lane%16
    const int orow_base = tileM + m0 + hi * 8;
    const int oc0 = tileN + n0 + hl;
    const int oc1 = oc0 + 16;
    float den0 = 1.0f, den1 = 1.0f;
    if (colDen) {
        const float* cd = colDen + (size_t)bv * cBv;
        den0 = 1.0f / cd[oc0];
        den1 = 1.0f / cd[oc1];
    }
    #pragma unroll
    for (int r = 0; r < 8; ++r) {
        const int orow = orow_base + r;
        const float b = bias ? bias[orow] : 0.0f;
        Out[(size_t)orow * ldo + oc0] = (acc0[r] + b) * den0;
        Out[(size_t)orow * ldo + oc1] = (acc1[r] + b) * den1;
    }
}

// ---------------- K4: per-channel BN stats over (BV, N): mean, rsqrt(var+eps)
__global__ __launch_bounds__(256) void bn_stats(
    const float* __restrict__ y, size_t bvStride,
    float* __restrict__ mean, float* __restrict__ rstd)
{
    const int c = blockIdx.x;
    __shared__ float s1[256];
    __shared__ float s2[256];
    const int total = BV * Nt;
    float a1 = 0.0f, a2 = 0.0f;
    for (int i = threadIdx.x; i < total; i += 256) {
        const int bv = i / Nt;
        const int n  = i - bv * Nt;
        const float v = y[(size_t)bv * bvStride + (size_t)c * Nt + n];
        a1 += v; a2 += v * v;
    }
    s1[threadIdx.x] = a1; s2[threadIdx.x] = a2;
    __syncthreads();
    for (int s = 128; s > 0; s >>= 1) {
        if (threadIdx.x < s) { s1[threadIdx.x] += s1[threadIdx.x + s];
                               s2[threadIdx.x] += s2[threadIdx.x + s]; }
        __syncthreads();
    }
    if (threadIdx.x == 0) {
        const float m = s1[0] / (float)total;
        const float var = s2[0] / (float)total - m * m;
        mean[c] = m;
        rstd[c] = rsqrtf(var + BN_EPS);
    }
}

// ---------------- K5: fold branch BNs into w_1/b_1:  w1'[o][c]=w1[o][c]*s_c, b1'[o]=b1[o]+sum_c w1[o][c]*t_c
__global__ __launch_bounds__(256) void fold_w1(
    const float* __restrict__ w1, const float* __restrict__ b1,
    const float* __restrict__ g_ca, const float* __restrict__ be_ca,
    const float* __restrict__ g_ctx, const float* __restrict__ be_ctx,
    const float* __restrict__ meanF, const float* __restrict__ rstdF,
    float* __restrict__ w1p, float* __restrict__ b1p)
{
    const int o = blockIdx.x;
    __shared__ float red[256];
    float acc = 0.0f;
    for (int c = threadIdx.x; c < C2; c += 256) {
        const float g  = (c < Cc) ? g_ca[c]  : g_ctx[c - Cc];
        const float be = (c < Cc) ? be_ca[c] : be_ctx[c - Cc];
        const float s = g * rstdF[c];
        const float t = be - meanF[c] * s;
        const float w = w1[(size_t)o * C2 + c];
        w1p[(size_t)o * C2 + c] = w * s;
        acc += w * t;
    }
    red[threadIdx.x] = acc;
    __syncthreads();
    for (int s = 128; s > 0; s >>= 1) {
        if (threadIdx.x < s) red[threadIdx.x] += red[threadIdx.x + s];
        __syncthreads();
    }
    if (threadIdx.x == 0) b1p[o] = b1[o] + red[0];
}

// ---------------- K6: final BN + gamma*out + residual
__global__ __launch_bounds__(256) void final_kernel(
    const float* __restrict__ y1, const float* __restrict__ x,
    const float* __restrict__ mean1, const float* __restrict__ rstd1,
    const float* __restrict__ g1, const float* __restrict__ be1,
    const float* __restrict__ gamma, float* __restrict__ out)
{
    const size_t idx = (size_t)blockIdx.x * 256 + threadIdx.x;   // over BV*Cc*Nt
    const int c = (int)((idx / Nt) % Cc);
    const float y = y1[idx];
    const float bn = g1[c] * (y - mean1[c]) * rstd1[c] + be1[c];
    out[idx] = gamma[0] * bn + x[idx];
}

// ---------------- launch ----------------
extern "C" void kernel_launch(void* const* d_in, const int* in_sizes, int n_in,
                              void* d_out, int out_size, void* d_ws, size_t ws_size,
                              hipStream_t stream)
{
    const float* ctx     = (const float*)d_in[0];   // [BV, C, N]
    const float* dist    = (const float*)d_in[1];   // [BV, N, N]
    const float* w_value = (const float*)d_in[2];
    const float* b_value = (const float*)d_in[3];
    const float* w_ca    = (const float*)d_in[4];
    const float* b_ca    = (const float*)d_in[5];
    const float* g_ca    = (const float*)d_in[6];
    const float* be_ca   = (const float*)d_in[7];
    const float* w_ctx   = (const float*)d_in[8];
    const float* b_ctx   = (const float*)d_in[9];
    const float* g_ctx   = (const float*)d_in[10];
    const float* be_ctx  = (const float*)d_in[11];
    const float* w_1     = (const float*)d_in[12];
    const float* b_1     = (const float*)d_in[13];
    const float* g_1     = (const float*)d_in[14];
    const float* be_1    = (const float*)d_in[15];
    const float* gamma   = (const float*)d_in[16];
    const float* sigma   = (const float*)d_in[17];

    float* out_f   = (float*)d_out;                         // [BV,C,N] then [B,N,N]
    float* attnOut = out_f + (size_t)BV * Cc * Nt;

    // workspace layout (floats)
    float* ws = (float*)d_ws;
    float* attn    = ws;                                    // BV*N*N
    float* norm    = attn    + (size_t)BV * Nt * Nt;        // BV*N
    float* proj    = norm    + (size_t)BV * Nt;             // BV*C*N
    float* ca      = proj    + (size_t)BV * Cc * Nt;        // BV*C*N
    float* yfused  = ca      + (size_t)BV * Cc * Nt;        // BV*2C*N
    float* y1      = yfused  + (size_t)BV * C2 * Nt;        // BV*C*N
    float* meanF   = y1      + (size_t)BV * Cc * Nt;        // 2C
    float* rstdF   = meanF   + C2;                          // 2C
    float* mean1   = rstdF   + C2;                          // C
    float* rstd1   = mean1   + Cc;                          // C
    float* w1p     = rstd1   + Cc;                          // C*2C
    float* b1p     = w1p     + (size_t)Cc * C2;             // C
    float* partial = b1p     + Cc;                          // ROWCHUNKS*BV*N

    // 1) attention map + partial column sums (+ attention output slice)
    attn_kernel<<<dim3(Nt / 128, BV, ROWCHUNKS), 128, 0, stream>>>(
        dist, sigma, attn, partial, attnOut);

    // 2) deterministic normalizer reduction
    norm_fin<<<dim3((BV * Nt) / 256), 256, 0, stream>>>(partial, norm);

    const dim3 gemm_grid(Nt / BNT, Cc / BMT, BV);

    // 3) proj_value = w_value @ x + b_value
    wmma_gemm_f32<<<gemm_grid, 256, 0, stream>>>(
        w_value, 0, Cc, ctx, (size_t)Cc * Nt, Nt,
        b_value, nullptr, 0, proj, (size_t)Cc * Nt, Nt, Cc);

    // 4) ca = (proj @ attn) / normalizer     (the dominant 512x1920x1920 GEMM)
    wmma_gemm_f32<<<gemm_grid, 256, 0, stream>>>(
        proj, (size_t)Cc * Nt, Nt, attn, (size_t)Nt * Nt, Nt,
        nullptr, norm, (size_t)Nt, ca, (size_t)Cc * Nt, Nt, Nt);

    // 5) y_ca = w_ca @ ca + b_ca   -> fused channels [0,512)
    wmma_gemm_f32<<<gemm_grid, 256, 0, stream>>>(
        w_ca, 0, Cc, ca, (size_t)Cc * Nt, Nt,
        b_ca, nullptr, 0, yfused, (size_t)C2 * Nt, Nt, Cc);

    // 6) y_ctx = w_ctx @ x + b_ctx -> fused channels [512,1024)
    wmma_gemm_f32<<<gemm_grid, 256, 0, stream>>>(
        w_ctx, 0, Cc, ctx, (size_t)Cc * Nt, Nt,
        b_ctx, nullptr, 0, yfused + (size_t)Cc * Nt, (size_t)C2 * Nt, Nt, Cc);

    // 7) BN stats of both branches (2C channels), then fold into w_1/b_1
    bn_stats<<<C2, 256, 0, stream>>>(yfused, (size_t)C2 * Nt, meanF, rstdF);
    fold_w1<<<Cc, 256, 0, stream>>>(w_1, b_1, g_ca, be_ca, g_ctx, be_ctx,
                                    meanF, rstdF, w1p, b1p);

    // 8) y1 = w1' @ yfused + b1'   (K = 1024)
    wmma_gemm_f32<<<gemm_grid, 256, 0, stream>>>(
        w1p, 0, C2, yfused, (size_t)C2 * Nt, Nt,
        b1p, nullptr, 0, y1, (size_t)Cc * Nt, Nt, C2);

    // 9) final BN + residual epilogue
    bn_stats<<<Cc, 256, 0, stream>>>(y1, (size_t)Cc * Nt, mean1, rstd1);
    final_kernel<<<((size_t)BV * Cc * Nt) / 256, 256, 0, stream>>>(
        y1, ctx, mean1, rstd1, g_1, be_1, gamma, out_f);
}